// AttentionDownSample_31499290149323
// MI455X (gfx1250) — compile-verified
//
#include <hip/hip_runtime.h>
#include <math.h>

// fp32 WMMA fragments
typedef float v2f __attribute__((ext_vector_type(2)));
typedef float v4f __attribute__((ext_vector_type(4)));
typedef float v8f __attribute__((ext_vector_type(8)));

#define TILE_STRIDE 130   // 128 channels + 2 pad dwords (bank spread)
#define WSTRIDE 33        // 32 + 1 pad (kp/qp result regions)

// Geometry: fm [8,128,256,256], Wq/Wk [128,32], out [8,128,128,128]
// One block = 16 consecutive windows in one row of one batch image.
__global__ __launch_bounds__(128)
void attn_downsample_wmma(const float* __restrict__ fm,
                          const float* __restrict__ Wq,
                          const float* __restrict__ Wk,
                          float* __restrict__ out)
{
    // LDS: pixel tile (rows 0..63 = tokens P=t*16+w, rows 64..79 = window means)
    __shared__ __align__(16) float tile[80 * TILE_STRIDE]; // 41600 B
    // Weights, K-pair interleaved: sW[cp*32 + col] = {W[2cp][col], W[2cp+1][col]}
    __shared__ v2f sWk2[64 * 32];                          // 16384 B
    __shared__ v2f sWq2[64 * 32];                          // 16384 B
    __shared__ float skp[64 * WSTRIDE];                    // kp[t][w][r]  8448 B
    __shared__ float sqp[64 * WSTRIDE];                    // qp partials  8448 B
    __shared__ float sqk[64];                              // qk[w][t]

    const int tid = threadIdx.x;
    const int wt  = blockIdx.x;               // 0..8191
    const int ww0 = (wt & 7) * 16;            // window-col base (nW=128)
    const int hh  = (wt >> 3) & 127;          // window row
    const int b   = wt >> 10;                 // batch

    // ---------------- Phase 1: stage fm tile + weights into LDS ----------------
    {
        // q = j*128 + tid: per-thread (yl,xq) are loop-invariant; c = j*8 + tid/16
        const int c0  = tid >> 4;
        const int rem = tid & 15;
        const int yl  = rem >> 3;
        const int xq  = rem & 7;
        const float* gp = &fm[(long)b * 128 * 65536 + (long)c0 * 65536
                              + (hh * 2 + yl) * 256 + ww0 * 2 + xq * 4];
        // LDS rows for the 4 x-positions of this thread's quad (loop-invariant)
        int ldsb[4];
        #pragma unroll
        for (int e = 0; e < 4; ++e) {
            int xl = xq * 4 + e;
            int P  = ((yl << 1) + (xl & 1)) * 16 + (xl >> 1);  // P = t*16 + w
            ldsb[e] = P * TILE_STRIDE + c0;
        }
        #pragma unroll
        for (int j = 0; j < 16; ++j) {
            v4f v = __builtin_nontemporal_load((const v4f*)(gp + (long)j * 8 * 65536));
            tile[ldsb[0] + j * 8] = v.x;
            tile[ldsb[1] + j * 8] = v.y;
            tile[ldsb[2] + j * 8] = v.z;
            tile[ldsb[3] + j * 8] = v.w;
        }
    }
    {
        // idx = j*128 + tid: r and parity loop-invariant; pair index advances by 2
        const int r    = tid & 31;
        const int chalf= tid >> 5;            // 0..3 -> c = j*4 + chalf
        const int par  = chalf & 1;
        const int cp0  = chalf >> 1;
        float* sWkF = (float*)sWk2;
        float* sWqF = (float*)sWq2;
        const int dst0 = (cp0 * 32 + r) * 2 + par;
        #pragma unroll
        for (int j = 0; j < 32; ++j) {
            int src = j * 128 + tid;
            int dst = dst0 + j * 128;         // (+2 pairs)*32 cols*2 floats
            sWkF[dst] = Wk[src];
            sWqF[dst] = Wq[src];
        }
    }
    __syncthreads();

    // ---------------- Phase 2: window means -> tile rows 64..79 ----------------
    // 128 threads x 16 windows: c = tid, w = j; all-immediate DS offsets
    #pragma unroll
    for (int j = 0; j < 16; ++j) {
        float s = tile[j * TILE_STRIDE + tid] + tile[(16 + j) * TILE_STRIDE + tid]
                + tile[(32 + j) * TILE_STRIDE + tid] + tile[(48 + j) * TILE_STRIDE + tid];
        tile[(64 + j) * TILE_STRIDE + tid] = 0.25f * s;
    }
    __syncthreads();

    // ---------------- Phase 3: WMMA projections ----------------
    const int wave  = tid >> 5;               // 0..3
    const int lane  = tid & 31;
    const int mrow  = lane & 15;              // M row / N col within tile
    const int khalf = lane >> 4;              // 0: K={0,1}, 1: K={2,3}
    const int mhi   = khalf << 3;             // +8 M rows for upper half lanes

    // (a) kp for token t = wave, full K = 128, Cr = 32 (two N tiles)
    {
        v8f acc0 = {0.f,0.f,0.f,0.f,0.f,0.f,0.f,0.f};
        v8f acc1 = {0.f,0.f,0.f,0.f,0.f,0.f,0.f,0.f};
        const float* ap = &tile[(wave * 16 + mrow) * TILE_STRIDE + khalf * 2];
        const v2f*   bp = &sWk2[khalf * 32 + mrow];
        #pragma unroll
        for (int ks = 0; ks < 32; ++ks) {
            v2f a  = *(const v2f*)(ap + ks * 4);   // +16 B/step
            v2f b0 = bp[ks * 64];                  // +512 B/step
            v2f b1 = bp[ks * 64 + 16];
            acc0 = __builtin_amdgcn_wmma_f32_16x16x4_f32(false, a, false, b0,
                                                         (short)0, acc0, false, false);
            acc1 = __builtin_amdgcn_wmma_f32_16x16x4_f32(false, a, false, b1,
                                                         (short)0, acc1, false, false);
        }
        float* kdst = &skp[(wave * 16 + mhi) * WSTRIDE + mrow];
        #pragma unroll
        for (int v = 0; v < 8; ++v) {
            kdst[v * WSTRIDE]      = acc0[v];
            kdst[v * WSTRIDE + 16] = acc1[v];
        }
    }

    // (b) qp partial over K quarter [wave*32, wave*32+32)
    {
        v8f acc0 = {0.f,0.f,0.f,0.f,0.f,0.f,0.f,0.f};
        v8f acc1 = {0.f,0.f,0.f,0.f,0.f,0.f,0.f,0.f};
        const float* ap = &tile[(64 + mrow) * TILE_STRIDE + wave * 32 + khalf * 2];
        const v2f*   bp = &sWq2[(wave * 16 + khalf) * 32 + mrow];
        #pragma unroll
        for (int ks = 0; ks < 8; ++ks) {
            v2f a  = *(const v2f*)(ap + ks * 4);
            v2f b0 = bp[ks * 64];
            v2f b1 = bp[ks * 64 + 16];
            acc0 = __builtin_amdgcn_wmma_f32_16x16x4_f32(false, a, false, b0,
                                                         (short)0, acc0, false, false);
            acc1 = __builtin_amdgcn_wmma_f32_16x16x4_f32(false, a, false, b1,
                                                         (short)0, acc1, false, false);
        }
        float* qdst = &sqp[(wave * 16 + mhi) * WSTRIDE + mrow];
        #pragma unroll
        for (int v = 0; v < 8; ++v) {
            qdst[v * WSTRIDE]      = acc0[v];
            qdst[v * WSTRIDE + 16] = acc1[v];
        }
    }
    __syncthreads();

    // ---------------- Phase 4: qk dots (scaled) ----------------
    const int w = tid & 15;
    const int p = tid >> 4;                   // 0..7
    if (p < 4) {
        const float* qp0 = &sqp[w * WSTRIDE];
        const float* kp0 = &skp[(p * 16 + w) * WSTRIDE];
        float acc = 0.f;
        #pragma unroll
        for (int r = 0; r < 32; ++r) {
            float q = qp0[r] + qp0[16 * WSTRIDE + r]
                    + qp0[32 * WSTRIDE + r] + qp0[48 * WSTRIDE + r];
            acc = fmaf(q, kp0[r], acc);
        }
        sqk[w * 4 + p] = acc * 0.17677669529663687f;   // * Cr^-0.5
    }
    __syncthreads();

    // ---------------- Phase 5: softmax + weighted sum + store ----------------
    float q0 = sqk[w * 4 + 0], q1 = sqk[w * 4 + 1];
    float q2 = sqk[w * 4 + 2], q3 = sqk[w * 4 + 3];
    float mx = fmaxf(fmaxf(q0, q1), fmaxf(q2, q3));
    float e0 = __expf(q0 - mx), e1 = __expf(q1 - mx);
    float e2 = __expf(q2 - mx), e3 = __expf(q3 - mx);
    float inv = 1.f / (e0 + e1 + e2 + e3);
    float a0 = e0 * inv, a1 = e1 * inv, a2 = e2 * inv, a3 = e3 * inv;

    const float* t0 = &tile[w * TILE_STRIDE + p * 16];
    float* op = &out[(((long)b * 128 + p * 16) * 128 + hh) * 128 + ww0 + w];
    #pragma unroll
    for (int i = 0; i < 16; ++i) {
        float o = a0 * t0[i]
                + a1 * t0[16 * TILE_STRIDE + i]
                + a2 * t0[32 * TILE_STRIDE + i]
                + a3 * t0[48 * TILE_STRIDE + i];
        __builtin_nontemporal_store(o, op + (long)i * 16384);  // coalesced over w
    }
}

extern "C" void kernel_launch(void* const* d_in, const int* in_sizes, int n_in,
                              void* d_out, int out_size, void* d_ws, size_t ws_size,
                              hipStream_t stream) {
    (void)in_sizes; (void)n_in; (void)out_size; (void)d_ws; (void)ws_size;
    const float* fm = (const float*)d_in[0];
    const float* Wq = (const float*)d_in[1];
    const float* Wk = (const float*)d_in[2];
    float* out = (float*)d_out;
    // 8 batches * 128 window rows * 8 window-col tiles = 8192 blocks
    dim3 grid(8192), block(128);
    hipLaunchKernelGGL(attn_downsample_wmma, grid, block, 0, stream, fm, Wq, Wk, out);
}